// DiscriminativeLoss_48009144434963
// MI455X (gfx1250) — compile-verified
//
#include <hip/hip_runtime.h>
#include <hip/hip_bf16.h>

typedef float v2f __attribute__((ext_vector_type(2)));
typedef float v8f __attribute__((ext_vector_type(8)));

#define NUM_CLUSTERS 32
#define DIM          16
#define INTRA_MARGIN 0.5f
#define INTER_MARGIN 1.5f
#define EPS          1e-8f

// Workspace layout (floats):
//   [0   .. 511 ]  sums[32][16]
//   [512 .. 543 ]  counts[32]
//   [544 .. 1055]  means[32][16]
//   [1056.. 1087]  per_cluster_hinge_sum[32]
#define WS_FLOATS 1088

__global__ void zero_ws_kernel(float* __restrict__ ws, int n) {
    int i = blockIdx.x * blockDim.x + threadIdx.x;
    if (i < n) ws[i] = 0.0f;
}

// ---------------------------------------------------------------------------
// Kernel 1: segment sums via V_WMMA_F32_16X16X4_F32 (fp32, exact).
//   sums(32x16) = OneHot(labels)^T (32xN) x F (Nx16)
// Guard-free main loop over full 4-point chunks; counts accumulated in
// registers straight from the one-hot A values; tail (<4 points) handled by
// block 0 with direct global atomics.
// ---------------------------------------------------------------------------
__global__ void seg_sum_wmma_kernel(const float* __restrict__ feats,
                                    const int*   __restrict__ labels,
                                    float* __restrict__ sums,    // [32*16]
                                    float* __restrict__ counts,  // [32]
                                    int n_points) {
    __shared__ float lds_sums[NUM_CLUSTERS * DIM];
    __shared__ float lds_counts[NUM_CLUSTERS];

    const int t = threadIdx.x;
    for (int i = t; i < NUM_CLUSTERS * DIM; i += blockDim.x) lds_sums[i] = 0.0f;
    if (t < NUM_CLUSTERS) lds_counts[t] = 0.0f;
    __syncthreads();

    const int lane            = t & 31;      // wave32
    const int wave            = t >> 5;
    const int waves_per_block = blockDim.x >> 5;
    const int global_wave     = blockIdx.x * waves_per_block + wave;
    const int n_waves         = gridDim.x * waves_per_block;

    const int sel = lane >> 4;   // 0: lanes 0-15 (K=0,1), 1: lanes 16-31 (K=2,3)
    const int col = lane & 15;   // A: cluster row (M); B: dim column (N)

    v8f acc_lo0 = {}, acc_lo1 = {};   // clusters  0..15  x dims 0..15
    v8f acc_hi0 = {}, acc_hi1 = {};   // clusters 16..31  x dims 0..15
    float cnt_lo = 0.0f, cnt_hi = 0.0f;

    const int n_full = n_points >> 2;   // guard-free full chunks

    auto chunk_body = [&](int c, v8f& alo, v8f& ahi) {
        const int p0 = c * 4 + 2 * sel;        // K = 2*sel
        const int l0 = labels[p0];
        const int l1 = labels[p0 + 1];
        // Prefetch one grid-stride ahead (speculative; dropped if invalid).
        __builtin_prefetch(&feats[(size_t)(p0 + 4 * n_waves) * DIM + col], 0, 1);
        // B 4x16: VGPR0 = rows K=0/K=2, VGPR1 = rows K=1/K=3.
        v2f B = { feats[(size_t)p0 * DIM + col],
                  feats[(size_t)(p0 + 1) * DIM + col] };
        // A 16x4 one-hot: lane row M=col; VGPR0=K(2*sel), VGPR1=K(2*sel+1).
        v2f A_lo, A_hi;
        A_lo[0] = (l0 == col)      ? 1.0f : 0.0f;
        A_lo[1] = (l1 == col)      ? 1.0f : 0.0f;
        A_hi[0] = (l0 == col + 16) ? 1.0f : 0.0f;
        A_hi[1] = (l1 == col + 16) ? 1.0f : 0.0f;
        // Counts fall out of the one-hot for free.
        cnt_lo += A_lo[0] + A_lo[1];
        cnt_hi += A_hi[0] + A_hi[1];
        alo = __builtin_amdgcn_wmma_f32_16x16x4_f32(
            false, A_lo, false, B, (short)0, alo, false, false);
        ahi = __builtin_amdgcn_wmma_f32_16x16x4_f32(
            false, A_hi, false, B, (short)0, ahi, false, false);
    };

    int c = global_wave;
    for (; c + n_waves < n_full; c += 2 * n_waves) {
        chunk_body(c,           acc_lo0, acc_hi0);
        chunk_body(c + n_waves, acc_lo1, acc_hi1);
    }
    for (; c < n_full; c += n_waves) {
        chunk_body(c, acc_lo0, acc_hi0);
    }

    v8f acc_lo = acc_lo0 + acc_lo1;
    v8f acc_hi = acc_hi0 + acc_hi1;

    // D layout: VGPR r -> cluster (r + 8*sel), dim = col.
    const int mbase = sel * 8;
#pragma unroll
    for (int r = 0; r < 8; ++r) {
        atomicAdd(&lds_sums[(mbase + r) * DIM + col],      acc_lo[r]);
        atomicAdd(&lds_sums[(16 + mbase + r) * DIM + col], acc_hi[r]);
    }
    atomicAdd(&lds_counts[col],      cnt_lo);
    atomicAdd(&lds_counts[col + 16], cnt_hi);
    __syncthreads();

    for (int i = t; i < NUM_CLUSTERS * DIM; i += blockDim.x) atomicAdd(&sums[i], lds_sums[i]);
    if (t < NUM_CLUSTERS) atomicAdd(&counts[t], lds_counts[t]);

    // Tail: points [n_full*4, n_points) -> direct global atomics from block 0.
    if (blockIdx.x == 0) {
        const int rem = n_points & 3;
        if (t < rem * DIM) {
            const int p = n_full * 4 + t / DIM;
            const int d = t % DIM;
            atomicAdd(&sums[labels[p] * DIM + d], feats[(size_t)p * DIM + d]);
        }
        if (t < rem) {
            atomicAdd(&counts[labels[n_full * 4 + t]], 1.0f);
        }
    }
}

// ---------------------------------------------------------------------------
// Kernel 2: means = sums / counts   (1 block, 512 threads)
// ---------------------------------------------------------------------------
__global__ void means_kernel(const float* __restrict__ sums,
                             const float* __restrict__ counts,
                             float* __restrict__ means) {
    const int t = threadIdx.x;           // 0..511
    means[t] = sums[t] / counts[t >> 4];
}

// ---------------------------------------------------------------------------
// Kernel 3: intra pass. One thread per point; means cached in LDS.
// hinge^2( ||f - mean[label] + eps|| - 0.5 ) segment-summed per cluster.
// ---------------------------------------------------------------------------
__global__ void intra_kernel(const float* __restrict__ feats,
                             const int*   __restrict__ labels,
                             const float* __restrict__ means,
                             float* __restrict__ per_cluster,  // [32]
                             int n_points) {
    __shared__ __align__(16) float lds_means[NUM_CLUSTERS * DIM];
    __shared__ float lds_acc[NUM_CLUSTERS];

    const int t = threadIdx.x;
    for (int i = t; i < NUM_CLUSTERS * DIM; i += blockDim.x) lds_means[i] = means[i];
    if (t < NUM_CLUSTERS) lds_acc[t] = 0.0f;
    __syncthreads();

    const int stride = gridDim.x * blockDim.x;
    for (int i = blockIdx.x * blockDim.x + t; i < n_points; i += stride) {
        __builtin_prefetch(&feats[(size_t)(i + stride) * DIM], 0, 1);
        const int lab = labels[i];
        const float4* f4 = (const float4*)(feats + (size_t)i * DIM);
        const float4* m4 = (const float4*)(lds_means + lab * DIM);
        float ss = 0.0f;
#pragma unroll
        for (int q = 0; q < 4; ++q) {
            float4 f = f4[q];
            float4 m = m4[q];
            float dx = f.x - m.x + EPS;
            float dy = f.y - m.y + EPS;
            float dz = f.z - m.z + EPS;
            float dw = f.w - m.w + EPS;
            ss += dx * dx + dy * dy + dz * dz + dw * dw;
        }
        const float dist = sqrtf(ss);
        const float h    = fmaxf(dist - INTRA_MARGIN, 0.0f);
        atomicAdd(&lds_acc[lab], h * h);
    }
    __syncthreads();
    if (t < NUM_CLUSTERS) atomicAdd(&per_cluster[t], lds_acc[t]);
}

// ---------------------------------------------------------------------------
// Kernel 4: finalize. 1 block, 1024 threads (one per (i,j) cluster pair).
// ---------------------------------------------------------------------------
__global__ void finalize_kernel(const float* __restrict__ means,
                                const float* __restrict__ counts,
                                const float* __restrict__ per_cluster,
                                float* __restrict__ out) {
    __shared__ __align__(16) float lm[NUM_CLUSTERS * DIM];
    __shared__ float s_inter, s_intra, s_reg;

    const int t = threadIdx.x;           // 0..1023
    if (t < NUM_CLUSTERS * DIM) lm[t] = means[t];
    if (t == 0) { s_inter = 0.0f; s_intra = 0.0f; s_reg = 0.0f; }
    __syncthreads();

    const int i = t >> 5;
    const int j = t & 31;
    if (i != j) {
        float ss = 0.0f;
#pragma unroll
        for (int d = 0; d < DIM; ++d) {
            float v = lm[i * DIM + d] - lm[j * DIM + d] + EPS;
            ss += v * v;
        }
        const float dist = sqrtf(ss);
        const float h    = fmaxf(2.0f * INTER_MARGIN - dist, 0.0f);
        atomicAdd(&s_inter, h * h);
    }
    if (t < NUM_CLUSTERS) {
        atomicAdd(&s_intra, per_cluster[t] / counts[t]);
        float ss = 0.0f;
#pragma unroll
        for (int d = 0; d < DIM; ++d) {
            float v = lm[t * DIM + d] + EPS;
            ss += v * v;
        }
        atomicAdd(&s_reg, sqrtf(ss));
    }
    __syncthreads();

    if (t == 0) {
        const float intra = s_intra / (float)NUM_CLUSTERS;
        const float inter = s_inter / (float)(NUM_CLUSTERS * (NUM_CLUSTERS - 1));
        const float reg   = (s_reg / (float)NUM_CLUSTERS) * 0.001f;
        out[0] = intra + inter + reg;   // weights: intra=1, inter=1, reg pre-scaled
        out[1] = intra;
        out[2] = inter;
        out[3] = reg;
    }
}

extern "C" void kernel_launch(void* const* d_in, const int* in_sizes, int n_in,
                              void* d_out, int out_size, void* d_ws, size_t ws_size,
                              hipStream_t stream) {
    const float* feats  = (const float*)d_in[0];
    const int*   labels = (const int*)d_in[1];
    const int n_points  = in_sizes[1];   // 2,000,000

    float* ws          = (float*)d_ws;
    float* sums        = ws;             // 512
    float* counts      = ws + 512;       // 32
    float* means       = ws + 544;       // 512
    float* per_cluster = ws + 1056;      // 32
    float* out         = (float*)d_out;

    zero_ws_kernel<<<(WS_FLOATS + 255) / 256, 256, 0, stream>>>(ws, WS_FLOATS);
    seg_sum_wmma_kernel<<<2048, 256, 0, stream>>>(feats, labels, sums, counts, n_points);
    means_kernel<<<1, NUM_CLUSTERS * DIM, 0, stream>>>(sums, counts, means);
    intra_kernel<<<2048, 256, 0, stream>>>(feats, labels, means, per_cluster, n_points);
    finalize_kernel<<<1, 1024, 0, stream>>>(means, counts, per_cluster, out);
}